// MolecularGNN_90314572300808
// MI455X (gfx1250) — compile-verified
//
#include <hip/hip_runtime.h>

#define N_NODES 100000
#define N_EDGES 3200000
#define N_GRAPHS 4096
#define BN_EPS 1e-5f

typedef __attribute__((ext_vector_type(2))) float v2f;
typedef __attribute__((ext_vector_type(8))) float v8f;

// ---------------- utility ----------------
__global__ void k_zero(float* __restrict__ p, int n) {
  int i = blockIdx.x * blockDim.x + threadIdx.x;
  if (i < n) p[i] = 0.0f;
}

// degree via fp32 atomics (matches reference segment_sum of ones)
__global__ void k_deg(const int* __restrict__ dst, float* __restrict__ deg, int E) {
  int i = blockIdx.x * blockDim.x + threadIdx.x;
  if (i < E) atomicAdd(&deg[dst[i]], 1.0f);
}

// in place: deg -> rsqrt(deg + 1)
__global__ void k_dinv(float* __restrict__ d, int n) {
  int i = blockIdx.x * blockDim.x + threadIdx.x;
  if (i < n) d[i] = rsqrtf(d[i] + 1.0f);
}

__global__ void k_norme(const int* __restrict__ src, const int* __restrict__ dst,
                        const float* __restrict__ dinv, float* __restrict__ norme, int E) {
  int i = blockIdx.x * blockDim.x + threadIdx.x;
  if (i < E) norme[i] = dinv[src[i]] * dinv[dst[i]];
}

// ---------------- WMMA f32 GEMM, register-blocked over N ----------------
// D[M, NT*16] = A[M,K] @ B[K, NT*16] (+bias)(+relu), V_WMMA_F32_16X16X4_F32.
// One wave32 per 16-row slab; the wave holds all NT accumulator tiles so each
// A fragment is loaded exactly once (A is the 51MB operand; B is L0-resident).
// A frag (16x4):  lanes 0-15 row M=lane, VGPR0/1 = K{0,1}; lanes 16-31 = K{2,3}.
// B frag (4x16):  VGPR j: lanes 0-15 -> K=j, N=lane; lanes 16-31 -> K=j+2.
// D (16x16 f32):  VGPR j: lanes 0-15 -> M=j, N=lane; lanes 16-31 -> M=j+8.
template <int NT>
__global__ void k_gemm_wmma(const float* __restrict__ A, const float* __restrict__ B,
                            float* __restrict__ D, const float* __restrict__ bias,
                            int M, int K, int relu) {
  constexpr int N = NT * 16;
  int wave = blockIdx.x * (blockDim.x >> 5) + (threadIdx.x >> 5);
  int lane = threadIdx.x & 31;
  if (wave >= (M >> 4)) return;              // uniform per wave: EXEC stays all-ones
  int m0 = wave << 4;
  int half = lane >> 4;                      // 0: lanes 0-15, 1: lanes 16-31
  int r = lane & 15;

  const float* arow  = A + (size_t)(m0 + r) * K + 2 * half;  // 8B-aligned (K,k even)
  const float* bbase = B + (size_t)(2 * half) * N + r;

  v8f acc[NT] = {};
  for (int k = 0; k < K; k += 4) {
    v2f a = *(const v2f*)(arow + k);         // K = k+2*half, k+2*half+1 (loaded once)
    const float* bk = bbase + (size_t)k * N;
    #pragma unroll
    for (int t = 0; t < NT; ++t) {
      v2f b;
      b.x = bk[t * 16];                      // K = k   + 2*half, col t*16+r
      b.y = bk[N + t * 16];                  // K = k+1 + 2*half
      acc[t] = __builtin_amdgcn_wmma_f32_16x16x4_f32(false, a, false, b,
                                                     (short)0, acc[t], false, false);
    }
  }

  float* drow = D + (size_t)(m0 + 8 * half) * N + r;
  #pragma unroll
  for (int t = 0; t < NT; ++t) {
    float bb = bias ? bias[t * 16 + r] : 0.0f;   // column fixed per lane
    #pragma unroll
    for (int j = 0; j < 8; ++j) {
      float v = acc[t][j] + bb;
      if (relu) v = fmaxf(v, 0.0f);
      drow[(size_t)j * N + t * 16] = v;
    }
  }
}

// agg[i,f] = xw[i,f] * dinv[i]^2 + bias[f]   (self-loop contribution + bias)
__global__ void k_agg_init(const float* __restrict__ xw, const float* __restrict__ dinv,
                           const float* __restrict__ bias, float* __restrict__ agg,
                           int n, int dout) {
  int idx = blockIdx.x * blockDim.x + threadIdx.x;
  if (idx >= n * dout) return;
  int i = idx / dout, f = idx % dout;
  float di = dinv[i];
  agg[idx] = xw[idx] * di * di + bias[f];
}

// one wave32 per edge: vectorized gather of xw[src], scaled atomic scatter into agg[dst].
// xw/agg/norme all fit in the 192MB L2, so this runs at L2 atomic bandwidth.
template <int DOUT>
__global__ void k_scatter(const float* __restrict__ xw, const int* __restrict__ src,
                          const int* __restrict__ dst, const float* __restrict__ norme,
                          float* __restrict__ agg, int E) {
  int wave = blockIdx.x * (blockDim.x >> 5) + (threadIdx.x >> 5);
  int lane = threadIdx.x & 31;
  if (wave >= E) return;
  int s = src[wave], d = dst[wave];
  float ne = norme[wave];
  constexpr int PER = DOUT / 32;  // 4 (128-dim) or 2 (64-dim)
  const float* srow = xw + (size_t)s * DOUT + lane * PER;
  float* drow = agg + (size_t)d * DOUT + lane * PER;
  if (PER == 4) {
    float4 v = *(const float4*)srow;          // global_load_b128
    atomicAdd(&drow[0], v.x * ne);
    atomicAdd(&drow[1], v.y * ne);
    atomicAdd(&drow[2], v.z * ne);
    atomicAdd(&drow[3], v.w * ne);
  } else {
    float2 v = *(const float2*)srow;
    atomicAdd(&drow[0], v.x * ne);
    atomicAdd(&drow[1], v.y * ne);
  }
}

// h = relu(g*(agg-m)*rsqrt(v+eps)+be)
__global__ void k_bnrelu(const float* __restrict__ agg, const float* __restrict__ g,
                         const float* __restrict__ be, const float* __restrict__ m,
                         const float* __restrict__ v, float* __restrict__ h,
                         int total, int dout) {
  int idx = blockIdx.x * blockDim.x + threadIdx.x;
  if (idx >= total) return;
  int f = idx % dout;
  float y = g[f] * (agg[idx] - m[f]) * rsqrtf(v[f] + BN_EPS) + be[f];
  h[idx] = fmaxf(y, 0.0f);
}

// global mean pool (sums + counts via atomics)
__global__ void k_pool(const float* __restrict__ h, const int* __restrict__ batch,
                       float* __restrict__ pooled, float* __restrict__ cnt, int n) {
  int idx = blockIdx.x * blockDim.x + threadIdx.x;
  if (idx >= n * 64) return;
  int i = idx >> 6, f = idx & 63;
  int g = batch[i];
  atomicAdd(&pooled[(size_t)g * 64 + f], h[idx]);
  if (f == 0) atomicAdd(&cnt[g], 1.0f);
}

__global__ void k_pooldiv(float* __restrict__ pooled, const float* __restrict__ cnt, int total) {
  int idx = blockIdx.x * blockDim.x + threadIdx.x;
  if (idx >= total) return;
  pooled[idx] /= fmaxf(cnt[idx >> 6], 1.0f);
}

// final 64 -> 2 projection
__global__ void k_head(const float* __restrict__ fc2, const float* __restrict__ ow,
                       const float* __restrict__ ob, float* __restrict__ out, int G) {
  int idx = blockIdx.x * blockDim.x + threadIdx.x;
  if (idx >= G * 2) return;
  int g = idx >> 1, c = idx & 1;
  const float* row = fc2 + (size_t)g * 64;
  float s = 0.0f;
  #pragma unroll
  for (int k = 0; k < 64; ++k) s += row[k] * ow[k * 2 + c];
  out[idx] = s + ob[c];
}

// ---------------- host ----------------
extern "C" void kernel_launch(void* const* d_in, const int* in_sizes, int n_in,
                              void* d_out, int out_size, void* d_ws, size_t ws_size,
                              hipStream_t stream) {
  (void)in_sizes; (void)n_in; (void)out_size; (void)ws_size;
  const float* x = (const float*)d_in[0];
  const int* edge_index = (const int*)d_in[1];
  const int* src = edge_index;
  const int* dst = edge_index + N_EDGES;
  const int* batch = (const int*)d_in[2];
  // d_in[3] = num_graphs scalar (compile-time constant here)
  const float *w[3], *b[3], *g[3], *be[3], *m[3], *v[3];
  for (int l = 0; l < 3; ++l) {
    w[l]  = (const float*)d_in[4 + 6 * l + 0];
    b[l]  = (const float*)d_in[4 + 6 * l + 1];
    g[l]  = (const float*)d_in[4 + 6 * l + 2];
    be[l] = (const float*)d_in[4 + 6 * l + 3];
    m[l]  = (const float*)d_in[4 + 6 * l + 4];
    v[l]  = (const float*)d_in[4 + 6 * l + 5];
  }
  const float* fw0 = (const float*)d_in[22];
  const float* fb0 = (const float*)d_in[23];
  const float* fw1 = (const float*)d_in[24];
  const float* fb1 = (const float*)d_in[25];
  const float* ow  = (const float*)d_in[26];
  const float* ob  = (const float*)d_in[27];
  float* out = (float*)d_out;

  // workspace carve-out (256B aligned regions)
  char* ws = (char*)d_ws;
  size_t off = 0;
  auto take = [&](size_t bytes) -> float* {
    char* p = ws + off;
    off = (off + bytes + 255) & ~(size_t)255;
    return (float*)p;
  };
  float* dinv   = take((size_t)N_NODES * 4);            // deg, then rsqrt(deg+1)
  float* norme  = take((size_t)N_EDGES * 4);
  float* xw     = take((size_t)N_NODES * 128 * 4);
  float* agg    = take((size_t)N_NODES * 128 * 4);
  float* h      = take((size_t)N_NODES * 128 * 4);
  float* pooled = take((size_t)N_GRAPHS * 64 * 4);      // cnt is contiguous after
  float* cnt    = take((size_t)N_GRAPHS * 4);
  float* fc1    = take((size_t)N_GRAPHS * 128 * 4);
  float* fc2    = take((size_t)N_GRAPHS * 64 * 4);

  const int T = 256;
  // gcn_norm
  k_zero<<<(N_NODES + T - 1) / T, T, 0, stream>>>(dinv, N_NODES);
  k_deg<<<(N_EDGES + T - 1) / T, T, 0, stream>>>(dst, dinv, N_EDGES);
  k_dinv<<<(N_NODES + T - 1) / T, T, 0, stream>>>(dinv, N_NODES);
  k_norme<<<(N_EDGES + T - 1) / T, T, 0, stream>>>(src, dst, dinv, norme, N_EDGES);

  const int dins[3]  = {36, 128, 128};
  const int douts[3] = {128, 128, 64};
  const int mw = N_NODES / 16;                 // 6250 row-slab waves
  const int gemmBlocks = (mw + 3) / 4;         // 4 waves / 128-thread block
  const float* hin = x;
  for (int l = 0; l < 3; ++l) {
    int din = dins[l], dout = douts[l];
    if (dout == 128)
      k_gemm_wmma<8><<<gemmBlocks, 128, 0, stream>>>(hin, w[l], xw, nullptr, N_NODES, din, 0);
    else
      k_gemm_wmma<4><<<gemmBlocks, 128, 0, stream>>>(hin, w[l], xw, nullptr, N_NODES, din, 0);
    int ne = N_NODES * dout;
    k_agg_init<<<(ne + T - 1) / T, T, 0, stream>>>(xw, dinv, b[l], agg, N_NODES, dout);
    int eblocks = (N_EDGES + 7) / 8;  // 8 waves (edges) per 256-thread block
    if (dout == 128)
      k_scatter<128><<<eblocks, 256, 0, stream>>>(xw, src, dst, norme, agg, N_EDGES);
    else
      k_scatter<64><<<eblocks, 256, 0, stream>>>(xw, src, dst, norme, agg, N_EDGES);
    k_bnrelu<<<(ne + T - 1) / T, T, 0, stream>>>(agg, g[l], be[l], m[l], v[l], h, ne, dout);
    hin = h;
  }

  // global mean pool (pooled and cnt are contiguous: one zero pass)
  k_zero<<<(N_GRAPHS * 65 + T - 1) / T, T, 0, stream>>>(pooled, N_GRAPHS * 65);
  k_pool<<<((size_t)N_NODES * 64 + T - 1) / T, T, 0, stream>>>(h, batch, pooled, cnt, N_NODES);
  k_pooldiv<<<(N_GRAPHS * 64 + T - 1) / T, T, 0, stream>>>(pooled, cnt, N_GRAPHS * 64);

  // FC head (WMMA with fused bias+ReLU)
  const int fw = N_GRAPHS / 16;                // 256 row-slab waves
  const int fcBlocks = (fw + 3) / 4;
  k_gemm_wmma<8><<<fcBlocks, 128, 0, stream>>>(pooled, fw0, fc1, fb0, N_GRAPHS, 64, 1);
  k_gemm_wmma<4><<<fcBlocks, 128, 0, stream>>>(fc1, fw1, fc2, fb1, N_GRAPHS, 128, 1);
  k_head<<<(N_GRAPHS * 2 + T - 1) / T, T, 0, stream>>>(fc2, ow, ob, out, N_GRAPHS);
}